// HistogramMatchingLoss_4363686773218
// MI455X (gfx1250) — compile-verified
//
#include <hip/hip_runtime.h>
#include <hip/hip_bf16.h>

typedef float vf4 __attribute__((ext_vector_type(4)));
typedef float v2f __attribute__((ext_vector_type(2)));
typedef float v8f __attribute__((ext_vector_type(8)));

#define NBINS 256
#define NCH   3
#define NBATCH 16
#define HW    (512 * 512)          // contiguous span with constant channel
#define SLABS (NBATCH * NCH)       // 48
#define SUBS  8                    // blocks per slab
#define CHUNK (HW / SUBS)          // 32768 floats per block
#define THREADS 256
#define ITERS (CHUNK / (THREADS * 4))  // 32 float4 iterations
#define PFD   4                    // prefetch distance (iterations)

// ---------------------------------------------------------------- init
__global__ void hml_init(unsigned* __restrict__ hist) {
    for (int i = threadIdx.x; i < 2 * NCH * NBINS; i += THREADS) hist[i] = 0u;
}

// ---------------------------------------------------------------- histogram
__global__ __launch_bounds__(THREADS) void hml_hist(const float* __restrict__ img1,
                                                    const float* __restrict__ img2,
                                                    unsigned* __restrict__ hist) {
    __shared__ unsigned lh[8][NBINS];          // per-wave private histograms
    const int tid  = threadIdx.x;
    const int wave = tid >> 5;                 // wave32

    for (int i = tid; i < 8 * NBINS; i += THREADS) ((unsigned*)lh)[i] = 0u;
    __syncthreads();

    const int img = blockIdx.y;                // 0 = img1, 1 = img2
    const float* src = img ? img2 : img1;
    const int slab = blockIdx.x / SUBS;        // (b*3 + c)
    const int sub  = blockIdx.x % SUBS;
    const int ch   = slab % NCH;
    const size_t base = (size_t)slab * HW + (size_t)sub * CHUNK;
    const vf4* p = (const vf4*)(src + base);   // 16B aligned (base % 32768 == 0)

    for (int i = 0; i < ITERS; ++i) {
        if (i + PFD < ITERS)
            __builtin_prefetch(&p[(i + PFD) * THREADS + tid], 0, 0); // global_prefetch_b8
        vf4 v = __builtin_nontemporal_load(&p[i * THREADS + tid]);   // b128 NT load
        #pragma unroll
        for (int k = 0; k < 4; ++k) {
            float x = v[k];
            if (x >= 0.0f && x <= 1.0f) {      // NaN excluded automatically
                int b = (int)floorf(x * 256.0f);
                b = b > 255 ? 255 : (b < 0 ? 0 : b);
                atomicAdd(&lh[wave][b], 1u);   // ds_add_u32
            }
        }
    }
    __syncthreads();

    unsigned s = 0;
    #pragma unroll
    for (int w = 0; w < 8; ++w) s += lh[w][tid];
    if (s) atomicAdd(&hist[(img * NCH + ch) * NBINS + tid], s);
}

// ---------------------------------------------------------------- finalize
// cumsum(d) over 256 bins as 16x16 matrix algebra on V_WMMA_F32_16X16X4_F32:
//   C = V*U + (Ls*V)*Ones,  U upper-tri ones, Ls strictly-lower ones.
__global__ __launch_bounds__(32) void hml_final(const unsigned* __restrict__ hist,
                                                float* __restrict__ out) {
    __shared__ float dld[NBINS];   // V, row-major: V[r][k] = dld[16r+k]
    __shared__ float qld[NBINS];   // Q = Ls*V, row-major
    const int tid = threadIdx.x;   // one wave32, EXEC all ones
    const int m   = tid & 15;      // A row / B,C,D column
    const int hi  = tid >> 4;      // K/M half select

    float total = 0.0f;
    for (int c = 0; c < NCH; ++c) {
        const unsigned* h1 = hist + c * NBINS;
        const unsigned* h2 = hist + (NCH + c) * NBINS;

        float t1 = 0.0f, t2 = 0.0f;
        #pragma unroll
        for (int j = 0; j < 8; ++j) {
            t1 += (float)h1[tid * 8 + j];
            t2 += (float)h2[tid * 8 + j];
        }
        #pragma unroll
        for (int s = 16; s; s >>= 1) {
            t1 += __shfl_xor(t1, s, 32);
            t2 += __shfl_xor(t2, s, 32);
        }
        const float i1 = 1.0f / t1, i2 = 1.0f / t2;
        #pragma unroll
        for (int j = 0; j < 8; ++j) {
            int b = tid * 8 + j;
            dld[b] = (float)h1[b] * i1 - (float)h2[b] * i2;
        }
        __syncthreads();

        // P = V*U ; Q = Ls*V  (K=16 as 4 slices of K=4)
        v8f P = {}; v8f Q = {};
        #pragma unroll
        for (int s4 = 0; s4 < 4; ++s4) {
            const int k0 = s4 * 4 + 2 * hi;    // K index for vgpr0 of this lane
            v2f a, b, a2, b2;
            // A = V slice (16x4): A[m][k0], A[m][k0+1]
            a.x = dld[16 * m + k0];
            a.y = dld[16 * m + k0 + 1];
            // B = U slice (4x16): U[k][n] = (k <= n), n == m
            b.x = (k0     <= m) ? 1.0f : 0.0f;
            b.y = (k0 + 1 <= m) ? 1.0f : 0.0f;
            P = __builtin_amdgcn_wmma_f32_16x16x4_f32(false, a, false, b,
                                                      (short)0, P, false, false);
            // A = Ls slice: Ls[m][k] = (k < m)
            a2.x = (k0     < m) ? 1.0f : 0.0f;
            a2.y = (k0 + 1 < m) ? 1.0f : 0.0f;
            // B = V slice (4x16): V[k][n]
            b2.x = dld[16 * k0 + m];
            b2.y = dld[16 * (k0 + 1) + m];
            Q = __builtin_amdgcn_wmma_f32_16x16x4_f32(false, a2, false, b2,
                                                      (short)0, Q, false, false);
        }
        // Spill Q (C/D layout: row = j + 8*hi, col = m) to row-major LDS
        #pragma unroll
        for (int j = 0; j < 8; ++j) qld[(j + 8 * hi) * 16 + m] = Q[j];
        __syncthreads();

        // R = Q*Ones + P  -> full cumsum in C/D layout
        v8f R = P;
        #pragma unroll
        for (int s4 = 0; s4 < 4; ++s4) {
            const int k0 = s4 * 4 + 2 * hi;
            v2f a, b;
            a.x = qld[16 * m + k0];
            a.y = qld[16 * m + k0 + 1];
            b.x = 1.0f;
            b.y = 1.0f;
            R = __builtin_amdgcn_wmma_f32_16x16x4_f32(false, a, false, b,
                                                      (short)0, R, false, false);
        }
        float part = 0.0f;
        #pragma unroll
        for (int j = 0; j < 8; ++j) part += fabsf(R[j]);
        #pragma unroll
        for (int s = 16; s; s >>= 1) part += __shfl_xor(part, s, 32);
        total += part;
        __syncthreads();   // dld/qld reused next channel
    }
    if (tid == 0) out[0] = total / 3.0f;
}

// ---------------------------------------------------------------- launch
extern "C" void kernel_launch(void* const* d_in, const int* in_sizes, int n_in,
                              void* d_out, int out_size, void* d_ws, size_t ws_size,
                              hipStream_t stream) {
    (void)in_sizes; (void)n_in; (void)out_size; (void)ws_size;
    const float* img1 = (const float*)d_in[0];
    const float* img2 = (const float*)d_in[1];
    float*    out  = (float*)d_out;
    unsigned* hist = (unsigned*)d_ws;          // uint32 hist[2][3][256] = 6 KB

    hml_init<<<1, THREADS, 0, stream>>>(hist);
    dim3 grid(SLABS * SUBS, 2);                // 384 blocks x 2 images
    hml_hist<<<grid, THREADS, 0, stream>>>(img1, img2, hist);
    hml_final<<<1, 32, 0, stream>>>(hist, out);
}